// intra_domain_encoder_50216757624949
// MI455X (gfx1250) — compile-verified
//
#include <hip/hip_runtime.h>

// ---------------------------------------------------------------------------
// GCNConv + TransformerConv(4 heads, beta) + Conv1d pipeline for gfx1250.
// Dense GEMMs use V_WMMA_F32_16X16X4_F32 (full f32 precision; GEMM cost is
// negligible vs the edge scatter traffic). K=64 fully unrolled; all A/B
// fragments preloaded to registers, then 16 back-to-back WMMAs. Width NW is
// a template parameter so B loads lower to immediate-offset vmem ops.
// ---------------------------------------------------------------------------

#define DIMC  64
#define HEADS 4
#define HDC   256   // HEADS*DIMC

typedef float v2f __attribute__((ext_vector_type(2)));
typedef float v8f __attribute__((ext_vector_type(8)));

// ---------------- utility fills ----------------
__global__ void fill_f32(float* p, long n, float v) {
  long i = (long)blockIdx.x * blockDim.x + threadIdx.x;
  if (i < n) p[i] = v;
}
__global__ void fill_u32(unsigned* p, long n, unsigned v) {
  long i = (long)blockIdx.x * blockDim.x + threadIdx.x;
  if (i < n) p[i] = v;
}

// order-preserving float <-> u32 (for atomicMax on signed floats)
__device__ __forceinline__ unsigned encodeF(float f) {
  unsigned b = __float_as_uint(f);
  return (b & 0x80000000u) ? ~b : (b | 0x80000000u);
}
__device__ __forceinline__ float decodeF(unsigned u) {
  unsigned b = (u & 0x80000000u) ? (u & 0x7fffffffu) : ~u;
  return __uint_as_float(b);
}

// ---------------- GCN degree ----------------
__global__ void deg_accum(const float* __restrict__ ew, const int* __restrict__ dst,
                          float* __restrict__ deg, int E) {
  int e = blockIdx.x * blockDim.x + threadIdx.x;
  if (e < E) atomicAdd(&deg[dst[e]], ew[e]);
}
__global__ void dinv_kernel(float* deg, int N) {
  int i = blockIdx.x * blockDim.x + threadIdx.x;
  if (i < N) {
    float d = deg[i] + 1.0f;              // self-loop weight 1
    deg[i] = (d > 0.f) ? rsqrtf(d) : 0.f; // becomes dinv in-place
  }
}

// ---------------- WMMA f32 GEMM ----------------
// C(M x NW) = A(M x 64) @ B(64 x NW) (+bias). One wave per 16x16 tile.
// BT=1: B element (k,col) read from B[col*64 + k] (B given as NW x 64).
// ST=1: C element (row,col) stored to C[col*ldOut + row].
// ISA f32 fragment layouts (CDNA5 §7.12.2):
//   A 16x4:  v0 = {lanes0-15: K=kk | lanes16-31: K=kk+2}, v1 = {+1}, M=lane%16
//   B 4x16:  same K striping, N=lane%16
//   C 16x16: vgpr r = {lanes0-15: M=r, N=lane | lanes16-31: M=r+8, N=lane-16}
template <int BT, int ST, int NW>
__global__ void wmma_gemm64_f32(const float* __restrict__ A, const float* __restrict__ B,
                                const float* __restrict__ bias, float* __restrict__ C,
                                int M, int ldOut) {
  constexpr int K = 64;
  const int m0   = blockIdx.x * 16;
  const int n0   = blockIdx.y * 16;
  const int lane = threadIdx.x & 31;
  const int half = lane >> 4;   // 0: K pair {kk,kk+1}, 1: {kk+2,kk+3}
  const int l16  = lane & 15;
  const int rowA = m0 + l16;
  const int colB = n0 + l16;

  v8f acc = {};
  const float* Arow = A + (long)rowA * K + half * 2;
  const float* Bp   = BT ? (B + (long)colB * K + half * 2)
                         : (B + (long)(half * 2) * NW + colB);

  if (m0 + 16 <= M) {
    // ---- full tile: preload all fragments (immediate offsets), then 16 WMMAs
    v2f aR[16], bR[16];
#pragma unroll
    for (int i = 0; i < 16; ++i) {
      const int kk = i * 4;
      aR[i] = *(const v2f*)(Arow + kk);
      if (BT) {
        bR[i] = *(const v2f*)(Bp + kk);
      } else {
        bR[i].x = Bp[kk * NW];
        bR[i].y = Bp[(kk + 1) * NW];
      }
    }
#pragma unroll
    for (int i = 0; i < 16; ++i)
      acc = __builtin_amdgcn_wmma_f32_16x16x4_f32(false, aR[i], false, bR[i],
                                                  (short)0, acc, false, false);

    const float bv  = bias ? bias[n0 + l16] : 0.f;
    const int rBase = m0 + half * 8;
#pragma unroll
    for (int r = 0; r < 8; ++r) {
      const float val = acc[r] + bv;
      if (ST) C[(long)(n0 + l16) * ldOut + (rBase + r)] = val;
      else    C[(long)(rBase + r) * NW + (n0 + l16)]    = val;
    }
  } else {
    // ---- tail tile (not hit when M % 16 == 0): guarded ----
    const bool rowOK = (rowA < M);
#pragma unroll
    for (int kk = 0; kk < K; kk += 4) {
      v2f a = {0.f, 0.f};
      if (rowOK) a = *(const v2f*)(Arow + kk);
      v2f b;
      if (BT) {
        b = *(const v2f*)(Bp + kk);
      } else {
        b.x = Bp[kk * NW];
        b.y = Bp[(kk + 1) * NW];
      }
      acc = __builtin_amdgcn_wmma_f32_16x16x4_f32(false, a, false, b,
                                                  (short)0, acc, false, false);
    }
    const float bv  = bias ? bias[n0 + l16] : 0.f;
    const int rBase = m0 + half * 8;
#pragma unroll
    for (int r = 0; r < 8; ++r) {
      const int row = rBase + r;
      if (row < M) {
        const float val = acc[r] + bv;
        if (ST) C[(long)(n0 + l16) * ldOut + row] = val;
        else    C[(long)row * NW + (n0 + l16)]    = val;
      }
    }
  }
}

// ---------------- GCN edge scatter + node finalize ----------------
__global__ void gcn_scatter(const float* __restrict__ xw, const float* __restrict__ ew,
                            const float* __restrict__ dinv, const int* __restrict__ src,
                            const int* __restrict__ dst, float* __restrict__ agg, int E) {
  long idx = (long)blockIdx.x * blockDim.x + threadIdx.x;
  if (idx >= (long)E * DIMC) return;
  int e = (int)(idx >> 6), c = (int)(idx & 63);
  int s = src[e], d = dst[e];
  float nrm = dinv[s] * ew[e] * dinv[d];
  atomicAdd(&agg[(long)d * DIMC + c], nrm * xw[(long)s * DIMC + c]);
}
__global__ void fgcn_kernel(const float* __restrict__ agg, const float* __restrict__ xw,
                            const float* __restrict__ dinv, const float* __restrict__ gcn_b,
                            float* __restrict__ fgcn, int N) {
  long idx = (long)blockIdx.x * blockDim.x + threadIdx.x;
  if (idx >= (long)N * DIMC) return;
  int i = (int)(idx >> 6), c = (int)(idx & 63);
  float di = dinv[i];
  float v = agg[idx] + di * di * xw[idx] + gcn_b[c];
  fgcn[idx] = fmaxf(v, 0.f);
}

// ---------------- attention: logits + segment max ----------------
__global__ void logits_kernel(const float* __restrict__ q, const float* __restrict__ k,
                              const int* __restrict__ src, const int* __restrict__ dst,
                              float* __restrict__ logits, unsigned* __restrict__ m_u, int E) {
  int wid  = blockIdx.x * (blockDim.x >> 5) + (threadIdx.x >> 5);
  int lane = threadIdx.x & 31;
  if (wid >= E) return;
  int s = src[wid], d = dst[wid];
  int h  = lane >> 3;
  int c0 = (lane & 7) * 8;
  const float* qp = q + (long)d * HDC + h * DIMC + c0;
  const float* kp = k + (long)s * HDC + h * DIMC + c0;
  float sum = 0.f;
#pragma unroll
  for (int j = 0; j < 8; ++j) sum += qp[j] * kp[j];
  sum += __shfl_xor(sum, 1);
  sum += __shfl_xor(sum, 2);
  sum += __shfl_xor(sum, 4);
  if ((lane & 7) == 0) {
    float lg = sum * 0.125f;  // 1/sqrt(64)
    logits[(long)wid * HEADS + h] = lg;
    atomicMax(&m_u[(long)d * HEADS + h], encodeF(lg));
  }
}

// ---------------- exp + segment sum (logits becomes ex in-place) ------------
__global__ void ex_kernel(float* __restrict__ logits, const unsigned* __restrict__ m_u,
                          const int* __restrict__ dst, float* __restrict__ z, int E) {
  long idx = (long)blockIdx.x * blockDim.x + threadIdx.x;
  if (idx >= (long)E * HEADS) return;
  int e = (int)(idx >> 2), h = (int)(idx & 3);
  int d = dst[e];
  float mf  = decodeF(m_u[(long)d * HEADS + h]);
  float exv = expf(logits[idx] - mf);
  logits[idx] = exv;
  atomicAdd(&z[(long)d * HEADS + h], exv);
}

// ---------------- unnormalized value scatter: S[dst] += ex * v[src] ---------
__global__ void scatter_v(const float* __restrict__ ex, const float* __restrict__ v,
                          const int* __restrict__ src, const int* __restrict__ dst,
                          float* __restrict__ S) {
  int e = blockIdx.x;
  int t = threadIdx.x;           // 0..255 = h*64 + c
  int s = src[e], d = dst[e];
  int h = t >> 6;
  float a = ex[(long)e * HEADS + h];
  atomicAdd(&S[(long)d * HDC + t], a * v[(long)s * HDC + t]);
}

// ---------------- combine: head mean, beta gate, relu -> f_tf --------------
__global__ void combine_kernel(const float* __restrict__ S, const float* __restrict__ z,
                               const float* __restrict__ xr, const float* __restrict__ wbeta,
                               float* __restrict__ ftf, int N) {
  int node = blockIdx.x * (blockDim.x >> 5) + (threadIdx.x >> 5);
  int lane = threadIdx.x & 31;
  if (node >= N) return;
  float o[2], x[2];
  float p = 0.f;
#pragma unroll
  for (int j = 0; j < 2; ++j) {
    int c = lane + j * 32;
    float acc = 0.f;
#pragma unroll
    for (int h = 0; h < HEADS; ++h) {
      float zh = z[(long)node * HEADS + h];
      float sv = S[(long)node * HDC + h * DIMC + c];
      acc += (zh > 0.f) ? (sv / zh) : 0.f;   // empty segments -> 0
    }
    o[j] = acc * 0.25f;                      // head mean
    x[j] = xr[(long)node * DIMC + c];
    p += o[j] * wbeta[c] + x[j] * wbeta[DIMC + c] + (o[j] - x[j]) * wbeta[2 * DIMC + c];
  }
  p += __shfl_xor(p, 1);
  p += __shfl_xor(p, 2);
  p += __shfl_xor(p, 4);
  p += __shfl_xor(p, 8);
  p += __shfl_xor(p, 16);
  float beta = 1.f / (1.f + expf(-p));
#pragma unroll
  for (int j = 0; j < 2; ++j) {
    int c = lane + j * 32;
    float val = beta * x[j] + (1.f - beta) * o[j];
    ftf[(long)node * DIMC + c] = fmaxf(val, 0.f);
  }
}

// ---------------------------------------------------------------------------
extern "C" void kernel_launch(void* const* d_in, const int* in_sizes, int n_in,
                              void* d_out, int out_size, void* d_ws, size_t ws_size,
                              hipStream_t stream) {
  (void)n_in; (void)out_size; (void)ws_size;
  const int N = in_sizes[0] / DIMC;
  const int E = in_sizes[1];

  const float* f_mole = (const float*)d_in[0];
  const float* ew     = (const float*)d_in[1];
  const float* gcn_w  = (const float*)d_in[2];
  const float* gcn_b  = (const float*)d_in[3];
  const float* wq     = (const float*)d_in[4];
  const float* bq     = (const float*)d_in[5];
  const float* wk     = (const float*)d_in[6];
  const float* bk     = (const float*)d_in[7];
  const float* wv     = (const float*)d_in[8];
  const float* bv     = (const float*)d_in[9];
  const float* wskip  = (const float*)d_in[10];
  const float* bskip  = (const float*)d_in[11];
  const float* wbeta  = (const float*)d_in[12];
  const float* conv_w = (const float*)d_in[13];
  const float* conv_b = (const float*)d_in[14];
  const int*   edges  = (const int*)d_in[15];
  const int*   srcp   = edges;
  const int*   dstp   = edges + E;

  // workspace carve-up (floats)
  float* ws = (float*)d_ws;
  float* deg  = ws; ws += N;                 // becomes dinv in-place
  float* xw   = ws; ws += (long)N * DIMC;
  float* agg  = ws; ws += (long)N * DIMC;
  float* fgcn = ws; ws += (long)N * DIMC;
  float* q    = ws; ws += (long)N * HDC;
  float* kb   = ws; ws += (long)N * HDC;
  float* vb   = ws; ws += (long)N * HDC;
  float* xr   = ws; ws += (long)N * DIMC;
  float* lg   = ws; ws += (long)E * HEADS;   // logits, then ex in-place
  unsigned* mu = (unsigned*)ws; ws += (long)N * HEADS;
  float* z    = ws; ws += (long)N * HEADS;
  float* S    = ws; ws += (long)N * HDC;
  float* ftf  = ws; ws += (long)N * DIMC;

  auto cdiv = [](long a, long b) { return (int)((a + b - 1) / b); };

  // init accumulators
  fill_f32<<<cdiv(N, 256), 256, 0, stream>>>(deg, N, 0.f);
  fill_f32<<<cdiv((long)N * DIMC, 256), 256, 0, stream>>>(agg, (long)N * DIMC, 0.f);
  fill_f32<<<cdiv((long)N * HEADS, 256), 256, 0, stream>>>(z, (long)N * HEADS, 0.f);
  fill_f32<<<cdiv((long)N * HDC, 256), 256, 0, stream>>>(S, (long)N * HDC, 0.f);
  fill_u32<<<cdiv((long)N * HEADS, 256), 256, 0, stream>>>(mu, (long)N * HEADS, 0u);

  // GCN degree + normalization
  deg_accum<<<cdiv(E, 256), 256, 0, stream>>>(ew, dstp, deg, E);
  dinv_kernel<<<cdiv(N, 256), 256, 0, stream>>>(deg, N);

  // xw = f_mole @ gcn_w
  wmma_gemm64_f32<0, 0, DIMC><<<dim3(cdiv(N, 16), DIMC / 16), 32, 0, stream>>>(
      f_mole, gcn_w, nullptr, xw, N, 0);

  // agg[dst] += norm * xw[src]; finalize f_gcn
  gcn_scatter<<<cdiv((long)E * DIMC, 256), 256, 0, stream>>>(xw, ew, deg, srcp, dstp, agg, E);
  fgcn_kernel<<<cdiv((long)N * DIMC, 256), 256, 0, stream>>>(agg, xw, deg, gcn_b, fgcn, N);

  // q / k / v / skip projections
  wmma_gemm64_f32<0, 0, HDC><<<dim3(cdiv(N, 16), HDC / 16), 32, 0, stream>>>(
      fgcn, wq, bq, q, N, 0);
  wmma_gemm64_f32<0, 0, HDC><<<dim3(cdiv(N, 16), HDC / 16), 32, 0, stream>>>(
      fgcn, wk, bk, kb, N, 0);
  wmma_gemm64_f32<0, 0, HDC><<<dim3(cdiv(N, 16), HDC / 16), 32, 0, stream>>>(
      fgcn, wv, bv, vb, N, 0);
  wmma_gemm64_f32<0, 0, DIMC><<<dim3(cdiv(N, 16), DIMC / 16), 32, 0, stream>>>(
      fgcn, wskip, bskip, xr, N, 0);

  // attention softmax over incoming edges
  logits_kernel<<<cdiv(E, 8), 256, 0, stream>>>(q, kb, srcp, dstp, lg, mu, E);
  ex_kernel<<<cdiv((long)E * HEADS, 256), 256, 0, stream>>>(lg, mu, dstp, z, E);
  scatter_v<<<E, 256, 0, stream>>>(lg, vb, srcp, dstp, S);

  // head mean + beta gate + relu
  combine_kernel<<<cdiv(N, 8), 256, 0, stream>>>(S, z, xr, wbeta, ftf, N);

  // conv: out[o, n] = sum_k conv_w[o,k] * ftf[n,k] + conv_b[o] -> d_out[o*N+n]
  wmma_gemm64_f32<1, 1, DIMC><<<dim3(cdiv(N, 16), DIMC / 16), 32, 0, stream>>>(
      ftf, conv_w, conv_b, (float*)d_out, N, N);
}